// ObjectDetector_41120016892334
// MI455X (gfx1250) — compile-verified
//
#include <hip/hip_runtime.h>
#include <hip/hip_bf16.h>
#include <math.h>

// Problem constants (fixed by the reference)
#define H_DIM   384
#define W_DIM   384
#define C_DIM   80
#define TILE_X  16
#define TILE_Y  8
#define TILE_W  (TILE_X + 2)   // staged cols
#define TILE_H  (TILE_Y + 2)   // staged rows
#define KDET    512
#define NBINS   4096
#define CBUF_CAP 8192
#define LOGIT_TH (-2.1972245773362196f)   // log(0.1/0.9): sigmoid(v)>0.1  <=>  v>this

// d_ws layout:
//   int[0]           candidate-collect counter
//   int[1]           threshold bin
//   int[64..64+4096) histogram
//   byte 32768:      ulong cbuf[8192]  (64 KB)
#define HIST_BASE   64
#define WS_CBUF_OFF 32768

#define ASG __attribute__((address_space(1)))
#define ASL __attribute__((address_space(3)))

typedef int v4i __attribute__((ext_vector_type(4)));

#if __has_builtin(__builtin_amdgcn_global_load_async_to_lds_b128)
#define ASYNC_COPY16(gsrc, ldst)                                                     \
  __builtin_amdgcn_global_load_async_to_lds_b128((ASG v4i*)(gsrc), (ASL v4i*)(ldst), 0, 0)
#define HAVE_ASYNC 1
#else
#define ASYNC_COPY16(gsrc, ldst) (*(float4*)(ldst) = *(const float4*)(gsrc))
#define HAVE_ASYNC 0
#endif

#if __has_builtin(__builtin_amdgcn_s_wait_asynccnt)
#define ASYNC_WAIT() __builtin_amdgcn_s_wait_asynccnt(0)
#elif HAVE_ASYNC
#define ASYNC_WAIT() asm volatile("s_wait_asynccnt 0" ::: "memory")
#else
#define ASYNC_WAIT()
#endif

__device__ __forceinline__ int bin_of(float v) {
  // candidates satisfy v > -2.1972; map [-2.25, 8.0] onto 4096 bins
  int b = (int)((v + 2.25f) * ((float)NBINS / 10.25f));
  if (b < 0) b = 0;
  if (b > NBINS - 1) b = NBINS - 1;
  return b;
}

__device__ __forceinline__ unsigned int ord_of_float(float f) {
  unsigned int u = __float_as_uint(f);
  return (u & 0x80000000u) ? ~u : (u | 0x80000000u);
}
__device__ __forceinline__ float float_of_ord(unsigned int o) {
  unsigned int u = (o & 0x80000000u) ? (o & 0x7FFFFFFFu) : ~o;
  return __uint_as_float(u);
}

__global__ void ws_init_kernel(int* __restrict__ ws_i) {
  int i = blockIdx.x * blockDim.x + threadIdx.x;
  if (i < HIST_BASE + NBINS) ws_i[i] = 0;
}

// 3x3 peak stencil over logits on an 8x16-pixel x 80-channel tile, staged through
// LDS with async global->LDS copies (10 rows incl. halo; 1.25x load redundancy).
// PHASE 0: build score histogram.  PHASE 1: collect candidates >= threshold bin.
template <int PHASE>
__global__ __launch_bounds__(256) void stencil_kernel(const float* __restrict__ hmap,
                                                      int* __restrict__ ws_i,
                                                      unsigned long long* __restrict__ cbuf) {
  __shared__ float lds[TILE_H * TILE_W * C_DIM];  // 10 x 18 x 80 f32 = 57.6 KB

  const int tid = threadIdx.x;
  const int bx  = blockIdx.x % (W_DIM / TILE_X);
  const int by  = blockIdx.x / (W_DIM / TILE_X);
  const int tx0 = bx * TILE_X;
  const int y0  = by * TILE_Y;

  const int colL = (tx0 == 0) ? 1 : 0;                               // first in-range LDS col
  const int colR = (tx0 + TILE_X == W_DIM) ? TILE_X : (TILE_X + 1);  // last in-range LDS col
  const int ncols = colR - colL + 1;
  const float NEG = __uint_as_float(0xff800000u);                    // -inf (SAME-pad identity)

  // Async-stage the rows (contiguous in NHWC: x-stride == C)
  for (int r = 0; r < TILE_H; ++r) {
    int gy = y0 - 1 + r;
    if (gy < 0 || gy >= H_DIM) continue;
    const float* gbase = hmap + ((size_t)gy * W_DIM + (size_t)(tx0 - 1 + colL)) * C_DIM;
    float* lbase = lds + (r * TILE_W + colL) * C_DIM;
    const int nflt = ncols * C_DIM;                                  // multiple of 4 (C=80)
    for (int j = tid; j * 4 < nflt; j += 256) {
      ASYNC_COPY16(gbase + j * 4, lbase + j * 4);
    }
  }
  // -inf fill for out-of-image halo (block-uniform branches)
  if (y0 == 0)
    for (int i = tid; i < TILE_W * C_DIM; i += 256) lds[i] = NEG;
  if (y0 + TILE_Y == H_DIM)
    for (int i = tid; i < TILE_W * C_DIM; i += 256)
      lds[(TILE_H - 1) * TILE_W * C_DIM + i] = NEG;
  if (tx0 == 0)
    for (int i = tid; i < TILE_H * C_DIM; i += 256)
      lds[(i / C_DIM) * TILE_W * C_DIM + (i % C_DIM)] = NEG;
  if (tx0 + TILE_X == W_DIM)
    for (int i = tid; i < TILE_H * C_DIM; i += 256)
      lds[((i / C_DIM) * TILE_W + (TILE_X + 1)) * C_DIM + (i % C_DIM)] = NEG;

  ASYNC_WAIT();
  __syncthreads();

  int tbin = 0;
  if (PHASE == 1) tbin = ws_i[1];

  // Flattened element space: e = pix*C + c, pix = ty*16 + txl. Stride 256 = 3*80 + 16.
  int c   = tid % C_DIM;
  int pix = tid / C_DIM;
  for (int e = tid; e < TILE_Y * TILE_X * C_DIM; e += 256) {
    const int txl = pix & (TILE_X - 1);
    const int ty  = pix >> 4;
    const float* cen = lds + ((ty + 1) * TILE_W + (txl + 1)) * C_DIM + c;
    const float v = cen[0];
    if (v > LOGIT_TH) {
      bool peak = true;
#pragma unroll
      for (int r = -1; r <= 1; ++r) {
#pragma unroll
        for (int dc = -1; dc <= 1; ++dc) {
          if (r == 0 && dc == 0) continue;
          peak = peak && (v >= cen[(r * TILE_W + dc) * C_DIM]);
        }
      }
      if (peak) {
        if (PHASE == 0) {
          atomicAdd(&ws_i[HIST_BASE + bin_of(v)], 1);
        } else if (bin_of(v) >= tbin) {
          int slot = atomicAdd(&ws_i[0], 1);
          if (slot < CBUF_CAP) {
            unsigned int fidx =
                ((unsigned)(y0 + ty) * W_DIM + (unsigned)(tx0 + txl)) * C_DIM + (unsigned)c;
            unsigned long long key =
                ((unsigned long long)ord_of_float(v) << 32) | (unsigned long long)(~fidx);
            cbuf[slot] = key;
          }
        }
      }
    }
    // advance (pix, c) by 256 elements
    pix += 3;
    c += 16;
    if (c >= C_DIM) { c -= C_DIM; ++pix; }
  }
}

// Find largest bin T whose suffix count reaches KDET.
__global__ void scan_kernel(int* __restrict__ ws_i) {
  if (threadIdx.x != 0 || blockIdx.x != 0) return;
  int acc = 0, tbin = 0;
  for (int b = NBINS - 1; b >= 0; --b) {
    acc += ws_i[HIST_BASE + b];
    if (acc >= KDET) { tbin = b; break; }
  }
  ws_i[1] = tbin;
}

// Bitonic sort (descending) of packed keys in LDS, then decode + write outputs.
__global__ __launch_bounds__(512) void sort_write_kernel(const float* __restrict__ rreg,
                                                         const float* __restrict__ bbox,
                                                         const int* __restrict__ ws_i,
                                                         const unsigned long long* __restrict__ cbuf,
                                                         float* __restrict__ out) {
  __shared__ unsigned long long sk[CBUF_CAP];  // 64 KB
  const int tid = threadIdx.x;
  int m = ws_i[0];
  if (m > CBUF_CAP) m = CBUF_CAP;
  for (int i = tid; i < CBUF_CAP; i += 512) sk[i] = (i < m) ? cbuf[i] : 0ull;
  __syncthreads();

  for (int k = 2; k <= CBUF_CAP; k <<= 1) {
    for (int j = k >> 1; j > 0; j >>= 1) {
      for (int i = tid; i < CBUF_CAP; i += 512) {
        int ixj = i ^ j;
        if (ixj > i) {
          unsigned long long a = sk[i], b = sk[ixj];
          bool desc = ((i & k) == 0);
          if (desc ? (a < b) : (a > b)) { sk[i] = b; sk[ixj] = a; }
        }
      }
      __syncthreads();
    }
  }

  if (tid < KDET) {
    float* out_cent = out;             // [512,2] (x,y)
    float* out_box  = out + 2 * KDET;  // [512,2]
    float* out_type = out + 4 * KDET;  // [512]
    float* out_scr  = out + 5 * KDET;  // [512]
    float* out_val  = out + 6 * KDET;  // [512]
    unsigned long long key = sk[tid];
    if (key == 0ull) {
      out_cent[2 * tid] = 0.0f; out_cent[2 * tid + 1] = 0.0f;
      out_box[2 * tid]  = 0.0f; out_box[2 * tid + 1]  = 0.0f;
      out_type[tid] = 0.0f;
      out_scr[tid]  = -1.0f;
      out_val[tid]  = 0.0f;
    } else {
      const float logit = float_of_ord((unsigned)(key >> 32));
      const unsigned fidx = ~(unsigned)key;
      const int c  = (int)(fidx % C_DIM);
      const unsigned pix = fidx / C_DIM;
      const int x  = (int)(pix % W_DIM);
      const int yy = (int)(pix / W_DIM);
      const float dy = rreg[(size_t)fidx * 2 + 0];
      const float dx = rreg[(size_t)fidx * 2 + 1];
      out_cent[2 * tid + 0] = (float)(int)rintf(((float)x  + dx) * 4.0f);
      out_cent[2 * tid + 1] = (float)(int)rintf(((float)yy + dy) * 4.0f);
      out_box[2 * tid + 0] = bbox[(size_t)pix * 2 + 0] * 4.0f;
      out_box[2 * tid + 1] = bbox[(size_t)pix * 2 + 1] * 4.0f;
      out_type[tid] = (float)c;
      out_scr[tid]  = 1.0f / (1.0f + expf(-logit));
      out_val[tid]  = 1.0f;
    }
  }
}

extern "C" void kernel_launch(void* const* d_in, const int* in_sizes, int n_in,
                              void* d_out, int out_size, void* d_ws, size_t ws_size,
                              hipStream_t stream) {
  const float* hmap = (const float*)d_in[0];
  const float* rreg = (const float*)d_in[1];
  const float* bbox = (const float*)d_in[2];
  float* out = (float*)d_out;
  int* ws_i = (int*)d_ws;
  unsigned long long* cbuf = (unsigned long long*)((char*)d_ws + WS_CBUF_OFF);

  const int n_init = HIST_BASE + NBINS;
  ws_init_kernel<<<(n_init + 255) / 256, 256, 0, stream>>>(ws_i);

  const int nblocks = (W_DIM / TILE_X) * (H_DIM / TILE_Y);  // 24 * 48 = 1152
  stencil_kernel<0><<<nblocks, 256, 0, stream>>>(hmap, ws_i, cbuf);
  scan_kernel<<<1, 32, 0, stream>>>(ws_i);
  stencil_kernel<1><<<nblocks, 256, 0, stream>>>(hmap, ws_i, cbuf);
  sort_write_kernel<<<1, 512, 0, stream>>>(rreg, bbox, ws_i, cbuf, out);
}